// GraphSAGE_73203422593459
// MI455X (gfx1250) — compile-verified
//
#include <hip/hip_runtime.h>
#include <hip/hip_bf16.h>
#include <math.h>

// ---------------------------------------------------------------------------
// GraphSAGE (max-pool aggregator), algebraically rewritten:
//   x[neigh] @ Wa  ==  (x @ Wa)[neigh]   (row-wise linear op commutes w/ gather)
// so each layer = dense GEMM (WMMA f32 16x16x4) + gather-max + concat GEMM.
// FLOPs drop from ~8.2 TFLOP to ~10 GFLOP; gathers hit L2 (Y table = 25.6 MB).
//
// Round-2 refinements (load-issue bound loop):
//  * weights pre-transposed once -> B fragment = single global_load_b64
//  * 2 column tiles per wave (A fragment reused) -> 1.5 vmem per wmma (was 3)
// ---------------------------------------------------------------------------

typedef float v2f __attribute__((ext_vector_type(2)));
typedef float v8f __attribute__((ext_vector_type(8)));

#define D 128

// WT[j,k] = W[k,j];  W is [K,128] row-major, WT is [128,K] row-major.
__global__ __launch_bounds__(256) void transpose_k(
    const float* __restrict__ W, float* __restrict__ WT, int K)
{
  int idx = blockIdx.x * 256 + threadIdx.x;   // over K*128 elements
  if (idx >= K * D) return;
  int k = idx >> 7;
  int j = idx & (D - 1);
  WT[(size_t)j * K + k] = W[(size_t)k * D + j];
}

// Y[n,0:128] = act( X0[n,:]@W[0:128,:] (+ X1[n,:]@W[128:256,:]) + bias )
// Block = 256 threads = 8 waves, covers 32 rows x 128 cols.
// Wave w: rows rowW..rowW+15 (rowW = row0 + 16*(w>=4)), cols [32*(w&3), +32)
// as two 16x16 WMMA tiles sharing the A fragment.
// WMMA f32 16x16x4 fragment layout (ISA 7.12.2):
//   A 16x4:  lane L holds A[M=L%16][k + 2*(L/16) + {0,1}]          -> v2f load
//   B 4x16:  lane L holds W[k + 2*(L/16) + {0,1}][col0 + L%16]
//            = WT[col0 + L%16][k + 2*(L/16) + {0,1}]               -> v2f load
//   C/D:     VGPR v, lane L -> row (v + 8*(L/16)), col (L%16)
template <bool RELU>
__global__ __launch_bounds__(256) void sage_gemm(
    const float* __restrict__ X0,
    const float* __restrict__ X1,     // nullptr => K=128, else K=256 (concat)
    const float* __restrict__ WT,     // [128,K] transposed weights
    const float* __restrict__ bias,   // [128]
    float* __restrict__ Y,            // [N,128]
    int N, int KT)                    // KT = WT row stride (128 or 256)
{
  const int lane = threadIdx.x & 31;
  const int wave = threadIdx.x >> 5;
  const int m    = lane & 15;
  const int kk   = (lane >> 4) << 1;            // 0 or 2
  const int rowW = (blockIdx.x << 5) + ((wave >> 2) << 4);
  const int c0   = (wave & 3) << 5;             // first col tile
  const int c1   = c0 + 16;                     // second col tile

  // clamp reads in the tail block so EXEC stays all-ones for WMMA
  const int rowRd = min(rowW + m, N - 1);

  const float* wt0 = WT + (size_t)(c0 + m) * KT;  // B fragments, tile 0
  const float* wt1 = WT + (size_t)(c1 + m) * KT;  // B fragments, tile 1

  v8f acc0 = {};
  v8f acc1 = {};

  const float* xrow = X0 + (size_t)rowRd * D;
#pragma unroll
  for (int k = 0; k < D; k += 4) {
    v2f a  = *(const v2f*)(xrow + k + kk);
    v2f b0 = *(const v2f*)(wt0 + k + kk);
    v2f b1 = *(const v2f*)(wt1 + k + kk);
    acc0 = __builtin_amdgcn_wmma_f32_16x16x4_f32(false, a, false, b0,
                                                 (short)0, acc0, false, false);
    acc1 = __builtin_amdgcn_wmma_f32_16x16x4_f32(false, a, false, b1,
                                                 (short)0, acc1, false, false);
  }
  if (X1) {
    const float* xrow1 = X1 + (size_t)rowRd * D;
#pragma unroll
    for (int k = 0; k < D; k += 4) {
      v2f a  = *(const v2f*)(xrow1 + k + kk);
      v2f b0 = *(const v2f*)(wt0 + D + k + kk);   // WT cols 128..255
      v2f b1 = *(const v2f*)(wt1 + D + k + kk);
      acc0 = __builtin_amdgcn_wmma_f32_16x16x4_f32(false, a, false, b0,
                                                   (short)0, acc0, false, false);
      acc1 = __builtin_amdgcn_wmma_f32_16x16x4_f32(false, a, false, b1,
                                                   (short)0, acc1, false, false);
    }
  }

  const float bv0   = bias[c0 + m];
  const float bv1   = bias[c1 + m];
  const int   rbase = rowW + ((lane >> 4) << 3);
#pragma unroll
  for (int v = 0; v < 8; ++v) {
    int r = rbase + v;
    if (r < N) {
      float v0 = acc0[v] + bv0;
      float v1 = acc1[v] + bv1;
      if (RELU) { v0 = fmaxf(v0, 0.0f); v1 = fmaxf(v1, 0.0f); }
      Y[(size_t)r * D + c0 + m] = v0;
      Y[(size_t)r * D + c1 + m] = v1;
    }
  }
}

// agg[n,:] = max_{s<S} Y[neigh[n,s],:]   -- one wave per node, float4 per lane.
__global__ __launch_bounds__(256) void gather_max_k(
    const float* __restrict__ Y, const int* __restrict__ neigh,
    float* __restrict__ agg, int N, int S)
{
  const int wave = threadIdx.x >> 5;
  const int lane = threadIdx.x & 31;
  const int node = blockIdx.x * 8 + wave;
  if (node >= N) return;

  const int* nb = neigh + (size_t)node * S;
  float4 m = make_float4(-3.402823466e38f, -3.402823466e38f,
                         -3.402823466e38f, -3.402823466e38f);
  for (int s = 0; s < S; ++s) {
    int idx = nb[s];
    float4 v = ((const float4*)(Y + (size_t)idx * D))[lane];
    m.x = fmaxf(m.x, v.x);
    m.y = fmaxf(m.y, v.y);
    m.z = fmaxf(m.z, v.z);
    m.w = fmaxf(m.w, v.w);
  }
  ((float4*)(agg + (size_t)node * D))[lane] = m;
}

// BatchNorm stats stage 1: per-block partial sum / sumsq per column (fixed order).
__global__ __launch_bounds__(128) void bn_partial_k(
    const float* __restrict__ X, float* __restrict__ psum,
    float* __restrict__ psq, int rowsPerBlock, int N)
{
  const int col = threadIdx.x;
  const int r0  = blockIdx.x * rowsPerBlock;
  const int r1  = min(r0 + rowsPerBlock, N);
  float s = 0.0f, q = 0.0f;
  for (int r = r0; r < r1; ++r) {
    float v = X[(size_t)r * D + col];
    s += v;
    q += v * v;
  }
  psum[blockIdx.x * D + col] = s;
  psq [blockIdx.x * D + col] = q;
}

// Stage 2: fold gamma/beta into per-column scale/shift. Deterministic serial sum.
__global__ __launch_bounds__(128) void bn_final_k(
    const float* __restrict__ psum, const float* __restrict__ psq,
    const float* __restrict__ gamma, const float* __restrict__ beta,
    float* __restrict__ scale, float* __restrict__ shift, int P, float invN)
{
  const int col = threadIdx.x;
  float s = 0.0f, q = 0.0f;
  for (int p = 0; p < P; ++p) {
    s += psum[p * D + col];
    q += psq [p * D + col];
  }
  const float mu   = s * invN;
  const float var  = q * invN - mu * mu;          // population var (jnp.var)
  const float rstd = 1.0f / sqrtf(var + 1e-5f);
  const float sc   = gamma[col] * rstd;
  scale[col] = sc;
  shift[col] = beta[col] - mu * sc;
}

// Fused BN affine + row L2-normalize. One wave per row, float4 per lane,
// wave32 shuffle-xor reduction for the row sum-of-squares.
__global__ __launch_bounds__(256) void bn_rownorm_k(
    const float* __restrict__ X, const float* __restrict__ scale,
    const float* __restrict__ shift, float* __restrict__ Yout, int N)
{
  const int wave = threadIdx.x >> 5;
  const int lane = threadIdx.x & 31;
  const int row  = blockIdx.x * 8 + wave;
  if (row >= N) return;

  float4 v  = ((const float4*)(X + (size_t)row * D))[lane];
  float4 sc = ((const float4*)scale)[lane];
  float4 sh = ((const float4*)shift)[lane];
  v.x = v.x * sc.x + sh.x;
  v.y = v.y * sc.y + sh.y;
  v.z = v.z * sc.z + sh.z;
  v.w = v.w * sc.w + sh.w;

  float ss = v.x * v.x + v.y * v.y + v.z * v.z + v.w * v.w;
#pragma unroll
  for (int off = 16; off > 0; off >>= 1) ss += __shfl_xor(ss, off, 32);

  const float inv = 1.0f / (sqrtf(ss) + 1e-6f);
  v.x *= inv; v.y *= inv; v.z *= inv; v.w *= inv;
  ((float4*)(Yout + (size_t)row * D))[lane] = v;
}

extern "C" void kernel_launch(void* const* d_in, const int* in_sizes, int n_in,
                              void* d_out, int out_size, void* d_ws, size_t ws_size,
                              hipStream_t stream)
{
  const float* features = (const float*)d_in[0];
  const int*   neigh0   = (const int*)  d_in[1];
  const int*   neigh1   = (const int*)  d_in[2];
  const float* Wa0      = (const float*)d_in[3];
  const float* ba0      = (const float*)d_in[4];
  const float* Wa1      = (const float*)d_in[5];
  const float* ba1      = (const float*)d_in[6];
  const float* W0       = (const float*)d_in[7];
  const float* b0       = (const float*)d_in[8];
  const float* W1       = (const float*)d_in[9];
  const float* b1       = (const float*)d_in[10];
  const float* gamma    = (const float*)d_in[11];
  const float* beta     = (const float*)d_in[12];
  float*       out      = (float*)d_out;

  const int N = in_sizes[0] / D;
  const int S = in_sizes[1] / N;

  // workspace layout (floats): 3 NxD slabs + BN partials + scale/shift + W^T
  const size_t slab = (size_t)N * D;
  float* bufA = (float*)d_ws;        // Y0 -> xn
  float* bufB = bufA + slab;         // agg0 -> Y1
  float* bufC = bufB + slab;         // x1  -> agg1
  const int P = 256;                 // BN partial blocks
  float* psum  = bufC + slab;
  float* psq   = psum + (size_t)P * D;
  float* scale = psq  + (size_t)P * D;
  float* shift = scale + D;
  float* Wa0T  = shift + D;                       // [128,128]
  float* Wa1T  = Wa0T + (size_t)D * D;            // [128,128]
  float* W0T   = Wa1T + (size_t)D * D;            // [128,256]
  float* W1T   = W0T  + (size_t)D * 2 * D;        // [128,256]

  const int gemmBlocks = (N + 31) / 32;
  const int waveBlocks = (N + 7) / 8;
  const int rowsPerBlk = (N + P - 1) / P;
  const int tr1 = (D * D + 255) / 256;            // transpose grid, K=128
  const int tr2 = (2 * D * D + 255) / 256;        // transpose grid, K=256

  // ---- one-time weight transposes (tiny) ----
  transpose_k<<<tr1, 256, 0, stream>>>(Wa0, Wa0T, D);
  transpose_k<<<tr1, 256, 0, stream>>>(Wa1, Wa1T, D);
  transpose_k<<<tr2, 256, 0, stream>>>(W0, W0T, 2 * D);
  transpose_k<<<tr2, 256, 0, stream>>>(W1, W1T, 2 * D);

  // ---- layer 0 ----
  // Y0 = relu(features @ Wa0 + ba0)
  sage_gemm<true><<<gemmBlocks, 256, 0, stream>>>(features, nullptr, Wa0T, ba0, bufA, N, D);
  // agg0 = max over neigh0 of Y0
  gather_max_k<<<waveBlocks, 256, 0, stream>>>(bufA, neigh0, bufB, N, S);
  // x1 = relu(concat(features, agg0) @ W0 + b0)   (relu fused)
  sage_gemm<true><<<gemmBlocks, 256, 0, stream>>>(features, bufB, W0T, b0, bufC, N, 2 * D);

  // ---- BatchNorm (batch stats) + row L2 normalize ----
  bn_partial_k<<<P, 128, 0, stream>>>(bufC, psum, psq, rowsPerBlk, N);
  bn_final_k<<<1, 128, 0, stream>>>(psum, psq, gamma, beta, scale, shift, P, 1.0f / (float)N);
  bn_rownorm_k<<<waveBlocks, 256, 0, stream>>>(bufC, scale, shift, bufA, N);  // xn -> bufA

  // ---- layer 1 ----
  // Y1 = relu(xn @ Wa1 + ba1)
  sage_gemm<true><<<gemmBlocks, 256, 0, stream>>>(bufA, nullptr, Wa1T, ba1, bufB, N, D);
  // agg1 = max over neigh1 of Y1
  gather_max_k<<<waveBlocks, 256, 0, stream>>>(bufB, neigh1, bufC, N, S);
  // out = concat(xn, agg1) @ W1 + b1   (no relu on last layer)
  sage_gemm<false><<<gemmBlocks, 256, 0, stream>>>(bufA, bufC, W1T, b1, out, N, 2 * D);
}